// SinkhornDistance_77738908057984
// MI455X (gfx1250) — compile-verified
//
#include <hip/hip_runtime.h>
#include <cstdint>
#include <cstddef>

// Sinkhorn: B=8, N=M=2048, eps=1e-3, 80 iterations.
// d_in[0]=mu [B,N] f32, d_in[1]=nu [B,M] f32, d_in[2]=cost [B,N,M] f32
// d_out = [ total_cost (B floats) | pi (B*N*M floats) ]

constexpr int   kB = 8;
constexpr int   kN = 2048;
constexpr int   kM = 2048;
constexpr int   kIters = 80;
constexpr float kEps = 0.001f;
constexpr float kLog2e = 1.4426950408889634f;
constexpr float kScale = kLog2e / kEps;            // (u - c) -> log2 domain
constexpr float kLn2 = 0.6931471805599453f;
constexpr float kLn2Eps = kLn2 * kEps;             // base-2 LSE back to nat, times eps
constexpr float kNegBig = -3.0e38f;

// ---- CDNA5 async global->LDS pipeline helpers (ASYNCcnt tracked) ----------

__device__ __forceinline__ uint32_t lds_addr32(const void* p) {
  // Generic pointers to LDS carry the workgroup-relative LDS offset in the
  // low 32 bits (ISA 10.2 aperture mapping).
  return (uint32_t)(uintptr_t)p;
}

// Stage = 8 "rows" x 32 floats (rows STRIDE floats apart) fetched with
// 2 x global_load_async_to_lds_b128 (32 lanes x 16B = 512B each).
//   glane = stage base + (lane>>3)*STRIDE + (lane&7)*4   (floats)
//   llane = LDS stage base + lane*16                      (bytes)
// Lane l's 16B land at LDS float index (l>>3)*32 + (l&7)*4, i.e. the
// [row*32 + col] layout the compute loops read.
template <int STRIDE_FLOATS>
__device__ __forceinline__ void async_cp_stage(const float* glane, uint32_t llane) {
  asm volatile("global_load_async_to_lds_b128 %0, %1, off"
               :
               : "v"(llane), "v"(glane)
               : "memory");
  asm volatile("global_load_async_to_lds_b128 %0, %1, off"
               :
               : "v"(llane + 512u), "v"(glane + (size_t)4 * STRIDE_FLOATS)
               : "memory");
}

#define WAIT_ASYNC_2() asm volatile("s_wait_asynccnt 2" ::: "memory")
#define WAIT_ASYNC_0() asm volatile("s_wait_asynccnt 0" ::: "memory")

__device__ __forceinline__ float fast_exp2(float x) { return __builtin_amdgcn_exp2f(x); }
__device__ __forceinline__ float fast_log2(float x) { return __builtin_amdgcn_logf(x); }

// ---- init: u = 1 ----------------------------------------------------------

__global__ void __launch_bounds__(256) sink_init(float* __restrict__ u) {
  u[blockIdx.x * 256 + threadIdx.x] = 1.0f;
}

// ---- v-update: v_j = eps*(log nu_j - LSE_i((u_i - c_ij)/eps)) -------------
// Block: batch b, 32 columns. Wave w reduces a 256-row chunk; LDS combine.

__global__ void __launch_bounds__(256) sink_vpass(const float* __restrict__ cost,
                                                  const float* __restrict__ u,
                                                  const float* __restrict__ nu,
                                                  float* __restrict__ v) {
  __shared__ float uk[kN];                           // u * kScale
  __shared__ alignas(16) float stage[8][2][256];     // [wave][buf][8 rows * 32 cols]
  __shared__ float mred[8][32];
  __shared__ float sred[8][32];

  const int b = blockIdx.y;
  const int j0 = blockIdx.x * 32;
  const int tid = threadIdx.x;
  const int w = tid >> 5;
  const int lane = tid & 31;

#pragma unroll
  for (int i = tid; i < kN; i += 256) uk[i] = u[b * kN + i] * kScale;
  __syncthreads();

  const int i0 = w * 256;  // this wave's row range
  // per-lane b128 source: row i0+(lane>>3), cols j0+(lane&7)*4
  const float* gl = cost + ((size_t)b * kN + i0 + (lane >> 3)) * kM + j0 + (lane & 7) * 4;
  const uint32_t l0 = lds_addr32(&stage[w][0][0]) + lane * 16u;
  const uint32_t l1 = lds_addr32(&stage[w][1][0]) + lane * 16u;

  float m = kNegBig, s = 0.0f;

  async_cp_stage<kM>(gl, l0);
  int buf = 0;
  for (int st = 0; st < 32; ++st) {        // 32 stages * 8 rows = 256 rows
    if (st + 1 < 32) {
      async_cp_stage<kM>(gl + (size_t)(st + 1) * 8 * kM, buf ? l0 : l1);
      WAIT_ASYNC_2();                      // previous stage complete (in-order)
    } else {
      WAIT_ASYNC_0();
    }
    const float* sb = &stage[w][buf][0];
    float x[8];
    float cm = kNegBig;
#pragma unroll
    for (int r = 0; r < 8; ++r) {
      float c = sb[r * 32 + lane];
      float uu = uk[i0 + st * 8 + r];
      x[r] = __builtin_fmaf(c, -kScale, uu);   // (u - c) * log2e/eps
      cm = fmaxf(cm, x[r]);
    }
    float mN = fmaxf(m, cm);
    s = s * fast_exp2(m - mN);
#pragma unroll
    for (int r = 0; r < 8; ++r) s += fast_exp2(x[r] - mN);
    m = mN;
    buf ^= 1;
  }

  mred[w][lane] = m;
  sred[w][lane] = s;
  __syncthreads();
  if (tid < 32) {
    float M2 = mred[0][tid], S2 = sred[0][tid];
#pragma unroll
    for (int ww = 1; ww < 8; ++ww) {
      float mo = mred[ww][tid], so = sred[ww][tid];
      float mN = fmaxf(M2, mo);
      S2 = S2 * fast_exp2(M2 - mN) + so * fast_exp2(mo - mN);
      M2 = mN;
    }
    const int j = j0 + tid;
    float lse2 = M2 + fast_log2(S2);                       // base-2 LSE
    float lognu = kLn2 * fast_log2(nu[b * kM + j] + 1e-8f);
    v[b * kM + j] = kEps * lognu - kLn2Eps * lse2;
  }
}

// ---- u-update: u_i = eps*(log mu_i - LSE_j((v_j - c_ij)/eps)) -------------
// Block: batch b, 8 rows; wave w owns one row, lanes stride columns by 32.

__global__ void __launch_bounds__(256) sink_upass(const float* __restrict__ cost,
                                                  const float* __restrict__ v,
                                                  const float* __restrict__ mu,
                                                  float* __restrict__ u) {
  __shared__ float vk[kM];                           // v * kScale
  __shared__ alignas(16) float stage[8][2][256];

  const int b = blockIdx.y;
  const int tid = threadIdx.x;
  const int w = tid >> 5;
  const int lane = tid & 31;
  const int i = blockIdx.x * 8 + w;

#pragma unroll
  for (int j = tid; j < kM; j += 256) vk[j] = v[b * kM + j] * kScale;
  __syncthreads();

  // contiguous stage: lane copies 16B at byte offset lane*16
  const float* gl = cost + ((size_t)b * kN + i) * kM + lane * 4;
  const uint32_t l0 = lds_addr32(&stage[w][0][0]) + lane * 16u;
  const uint32_t l1 = lds_addr32(&stage[w][1][0]) + lane * 16u;

  float m = kNegBig, s = 0.0f;

  async_cp_stage<32>(gl, l0);              // 256 cols per stage
  int buf = 0;
  for (int st = 0; st < 8; ++st) {         // 8 stages * 256 cols = 2048 cols
    if (st + 1 < 8) {
      async_cp_stage<32>(gl + (st + 1) * 256, buf ? l0 : l1);
      WAIT_ASYNC_2();
    } else {
      WAIT_ASYNC_0();
    }
    const float* sb = &stage[w][buf][0];
    float x[8];
    float cm = kNegBig;
#pragma unroll
    for (int k = 0; k < 8; ++k) {
      float c = sb[k * 32 + lane];
      float vv = vk[st * 256 + k * 32 + lane];
      x[k] = __builtin_fmaf(c, -kScale, vv);
      cm = fmaxf(cm, x[k]);
    }
    float mN = fmaxf(m, cm);
    s = s * fast_exp2(m - mN);
#pragma unroll
    for (int k = 0; k < 8; ++k) s += fast_exp2(x[k] - mN);
    m = mN;
    buf ^= 1;
  }

  // cross-lane (m, s) combine over 32 lanes
#pragma unroll
  for (int off = 16; off > 0; off >>= 1) {
    float mo = __shfl_xor(m, off, 32);
    float so = __shfl_xor(s, off, 32);
    float mN = fmaxf(m, mo);
    s = s * fast_exp2(m - mN) + so * fast_exp2(mo - mN);
    m = mN;
  }
  if (lane == 0) {
    float lse2 = m + fast_log2(s);
    float logmu = kLn2 * fast_log2(mu[b * kN + i] + 1e-8f);
    u[b * kN + i] = kEps * logmu - kLn2Eps * lse2;
  }
}

// ---- final: pi = exp((-c + u_i + v_j)/eps); partial dot(pi, cost) ---------

__global__ void __launch_bounds__(256) sink_final(const float* __restrict__ cost,
                                                  const float* __restrict__ u,
                                                  const float* __restrict__ v,
                                                  float* __restrict__ pi,
                                                  float* __restrict__ partial) {
  __shared__ float vk[kM];
  __shared__ alignas(16) float stage[8][2][256];
  __shared__ float wsum[8];

  const int b = blockIdx.y;
  const int tid = threadIdx.x;
  const int w = tid >> 5;
  const int lane = tid & 31;
  const int i = blockIdx.x * 8 + w;

#pragma unroll
  for (int j = tid; j < kM; j += 256) vk[j] = v[b * kM + j] * kScale;
  __syncthreads();

  const float uki = u[b * kN + i] * kScale;
  const size_t rowoff = ((size_t)b * kN + i) * kM;
  const float* gl = cost + rowoff + lane * 4;
  float* pbase = pi + rowoff + lane;
  const uint32_t l0 = lds_addr32(&stage[w][0][0]) + lane * 16u;
  const uint32_t l1 = lds_addr32(&stage[w][1][0]) + lane * 16u;

  float acc = 0.0f;
  async_cp_stage<32>(gl, l0);
  int buf = 0;
  for (int st = 0; st < 8; ++st) {
    if (st + 1 < 8) {
      async_cp_stage<32>(gl + (st + 1) * 256, buf ? l0 : l1);
      WAIT_ASYNC_2();
    } else {
      WAIT_ASYNC_0();
    }
    const float* sb = &stage[w][buf][0];
#pragma unroll
    for (int k = 0; k < 8; ++k) {
      float c = sb[k * 32 + lane];
      float p = fast_exp2(__builtin_fmaf(c, -kScale, uki + vk[st * 256 + k * 32 + lane]));
      __builtin_nontemporal_store(p, &pbase[st * 256 + k * 32]);  // write-once stream
      acc = __builtin_fmaf(p, c, acc);
    }
    buf ^= 1;
  }

#pragma unroll
  for (int off = 16; off > 0; off >>= 1) acc += __shfl_xor(acc, off, 32);
  if (lane == 0) wsum[w] = acc;
  __syncthreads();
  if (tid == 0) {
    float t = 0.0f;
#pragma unroll
    for (int ww = 0; ww < 8; ++ww) t += wsum[ww];
    partial[b * gridDim.x + blockIdx.x] = t;   // deterministic, no atomics
  }
}

__global__ void __launch_bounds__(256) sink_reduce(const float* __restrict__ partial,
                                                   float* __restrict__ out) {
  __shared__ float sh[256];
  const int b = blockIdx.x;
  const int t = threadIdx.x;
  sh[t] = partial[b * 256 + t];
  __syncthreads();
  for (int o = 128; o > 0; o >>= 1) {
    if (t < o) sh[t] += sh[t + o];
    __syncthreads();
  }
  if (t == 0) out[b] = sh[0];
}

// ---------------------------------------------------------------------------

extern "C" void kernel_launch(void* const* d_in, const int* in_sizes, int n_in,
                              void* d_out, int out_size, void* d_ws, size_t ws_size,
                              hipStream_t stream) {
  (void)in_sizes; (void)n_in; (void)out_size; (void)ws_size;
  const float* mu = (const float*)d_in[0];
  const float* nu = (const float*)d_in[1];
  const float* cost = (const float*)d_in[2];
  float* out = (float*)d_out;

  float* u = (float*)d_ws;                 // [B, N]
  float* v = u + (size_t)kB * kN;          // [B, M]
  float* partial = v + (size_t)kB * kM;    // [B * N/8 blocks] = 2048

  sink_init<<<(kB * kN) / 256, 256, 0, stream>>>(u);
  for (int it = 0; it < kIters; ++it) {
    sink_vpass<<<dim3(kM / 32, kB), 256, 0, stream>>>(cost, u, nu, v);
    sink_upass<<<dim3(kN / 8, kB), 256, 0, stream>>>(cost, v, mu, u);
  }
  sink_final<<<dim3(kN / 8, kB), 256, 0, stream>>>(cost, u, v, out + kB, partial);
  sink_reduce<<<kB, 256, 0, stream>>>(partial, out);
}